// Meta_upsample_3d_2_81930796138539
// MI455X (gfx1250) — compile-verified
//
#include <hip/hip_runtime.h>
#include <math.h>

typedef __attribute__((ext_vector_type(2))) float v2f;
typedef __attribute__((ext_vector_type(8))) float v8f;

#define CIN 31
#define HIN 128
#define WIN 128
#define HO 256
#define WO 256
#define NP (HO*WO)          // 65536 output pixels
#define NEXP 4
#define KCH 32

__device__ __forceinline__ v8f wmma_f32_16x16x4(v2f a, v2f b, v8f c) {
  // D = A(16x4 f32) * B(4x16 f32) + C(16x16 f32)
  return __builtin_amdgcn_wmma_f32_16x16x4_f32(false, a, false, b, (short)0, c, false, false);
}

// ---------------------------------------------------------------------------
// Kernel 1: routing softmax -> rsum, fold expert conv weights (we, wc)
// ---------------------------------------------------------------------------
__global__ void k_prep(const int* scale_p,
                       const float* we_in, const float* wc_in,
                       const float* lin1_w, const float* lin1_b,
                       const float* lin2_w, const float* lin2_b,
                       float* we_out, float* wc_out) {
  __shared__ float s_rsum[NEXP];
  int tid = threadIdx.x;
  if (tid == 0) {
    float fs = (float)(*scale_p);
    int s = (int)(fs + 0.5f);
    float rsum[NEXP];
    for (int e = 0; e < NEXP; ++e) rsum[e] = 0.f;
    for (int i = 0; i < s; ++i) {
      float fi = (float)i;
      float chv = (fi + 0.5f)/fs - floorf((fi + 0.5f)/fs + 0.001f) - 0.5f;
      float in0 = 1.0f/fs, in1 = chv;
      float h1[64];
      for (int f = 0; f < 64; ++f) {
        float z = lin1_w[f*2+0]*in0 + lin1_w[f*2+1]*in1 + lin1_b[f];
        h1[f] = fmaxf(z, 0.f);
      }
      float lg[NEXP];
      float mx = -1e30f;
      for (int o = 0; o < NEXP; ++o) {
        float z = lin2_b[o];
        for (int f = 0; f < 64; ++f) z += lin2_w[o*64+f]*h1[f];
        lg[o] = z; mx = fmaxf(mx, z);
      }
      float den = 0.f;
      for (int o = 0; o < NEXP; ++o) { lg[o] = __expf(lg[o]-mx); den += lg[o]; }
      for (int o = 0; o < NEXP; ++o) rsum[o] += lg[o]/den;
    }
    for (int e = 0; e < NEXP; ++e) s_rsum[e] = rsum[e];
  }
  __syncthreads();
  float r0 = s_rsum[0], r1 = s_rsum[1], r2 = s_rsum[2], r3 = s_rsum[3];
  // weight_expand (E,K,1,5,3,3) and weight_compress (E,1,K,5,3,3): both flat e*1440 + k*45 + tap
  for (int idx = tid; idx < KCH*45; idx += blockDim.x) {
    we_out[idx] = r0*we_in[idx] + r1*we_in[1440+idx] + r2*we_in[2880+idx] + r3*we_in[4320+idx];
    wc_out[idx] = r0*wc_in[idx] + r1*wc_in[1440+idx] + r2*wc_in[2880+idx] + r3*wc_in[4320+idx];
  }
}

// ---------------------------------------------------------------------------
// Kernel 2: per-pixel coordinate MLP. 64x64 layer via V_WMMA_F32_16X16X4_F32.
// 128 pixels/block (4 waves x 2 M-tiles of 16 pixels).
// ---------------------------------------------------------------------------
#define MLP_PX 128
__global__ void __launch_bounds__(128) k_mlp(const int* scale_p,
                      const float* w1, const float* b1,
                      const float* w2, const float* b2,
                      const float* rw, const float* rb,
                      const float* ow, const float* ob,
                      float* off0, float* off1, float* r2out) {
  __shared__ float s_e[MLP_PX*68];   // e1 then e2, pixel-major, stride 68 (bank spread)
  __shared__ float s_w2t[64*64];     // body_w2 transposed: [i][o]
  __shared__ float s_w1[64*3];
  __shared__ float s_b1[64];
  __shared__ float s_b2[64];
  __shared__ float s_rw[4*64];
  __shared__ float s_rb[4];
  __shared__ float s_ow[2*64];
  __shared__ float s_ob[2];
  int tid = threadIdx.x;
  for (int i = tid; i < 64*64; i += 128) {
    int o = i & 63, ii = i >> 6;
    s_w2t[ii*64 + o] = w2[o*64 + ii];
  }
  for (int i = tid; i < 64*3; i += 128) s_w1[i] = w1[i];
  for (int i = tid; i < 64;   i += 128) { s_b1[i] = b1[i]; s_b2[i] = b2[i]; }
  for (int i = tid; i < 4*64; i += 128) s_rw[i] = rw[i];
  for (int i = tid; i < 2*64; i += 128) s_ow[i] = ow[i];
  if (tid < 4) s_rb[tid] = rb[tid];
  if (tid < 2) s_ob[tid] = ob[tid];
  float fs = (float)(*scale_p);
  int p  = blockIdx.x * MLP_PX + tid;
  int py = p >> 8, px = p & 255;
  float chv = ((float)py + 0.5f)/fs - floorf(((float)py + 0.5f)/fs + 0.001f) - 0.5f;
  float cwv = ((float)px + 0.5f)/fs - floorf(((float)px + 0.5f)/fs + 0.001f) - 0.5f;
  float in0 = 1.0f/fs;
  __syncthreads();
  // layer 1: 3 -> 64, per-thread
  for (int f = 0; f < 64; ++f) {
    float z = s_w1[f*3+0]*in0 + s_w1[f*3+1]*chv + s_w1[f*3+2]*cwv + s_b1[f];
    s_e[tid*68 + f] = fmaxf(z, 0.f);
  }
  __syncthreads();
  // layer 2: 64x64 GEMM via WMMA f32 16x16x4
  int lane = tid & 31, wave = tid >> 5;
  int half = lane >> 4, mrow = lane & 15;
  v8f acc[2][4];
  #pragma unroll
  for (int mt = 0; mt < 2; ++mt)
    #pragma unroll
    for (int nt = 0; nt < 4; ++nt)
      acc[mt][nt] = {};
  #pragma unroll
  for (int mt = 0; mt < 2; ++mt) {
    int m0 = wave*32 + mt*16;
    const float* arow = &s_e[(m0 + mrow)*68 + 2*half];
    #pragma unroll
    for (int k = 0; k < 64; k += 4) {
      v2f a = *(const v2f*)(arow + k);
      #pragma unroll
      for (int nt = 0; nt < 4; ++nt) {
        v2f b;
        b.x = s_w2t[(k     + 2*half)*64 + nt*16 + mrow];
        b.y = s_w2t[(k + 1 + 2*half)*64 + nt*16 + mrow];
        acc[mt][nt] = wmma_f32_16x16x4(a, b, acc[mt][nt]);
      }
    }
  }
  __syncthreads();
  // bias + relu, scatter e2 back to LDS (C layout -> pixel-major)
  #pragma unroll
  for (int mt = 0; mt < 2; ++mt) {
    int m0 = wave*32 + mt*16;
    #pragma unroll
    for (int nt = 0; nt < 4; ++nt) {
      int n = nt*16 + mrow;
      #pragma unroll
      for (int v = 0; v < 8; ++v) {
        int m = m0 + v + 8*half;
        float val = acc[mt][nt][v] + s_b2[n];
        s_e[m*68 + n] = fmaxf(val, 0.f);
      }
    }
  }
  __syncthreads();
  // heads: offset (2) + routing2 sigmoid (4)
  float o0 = s_ob[0], o1 = s_ob[1];
  float rr0 = s_rb[0], rr1 = s_rb[1], rr2 = s_rb[2], rr3 = s_rb[3];
  for (int f = 0; f < 64; ++f) {
    float ev = s_e[tid*68 + f];
    o0  += s_ow[f]*ev;      o1  += s_ow[64+f]*ev;
    rr0 += s_rw[f]*ev;      rr1 += s_rw[64+f]*ev;
    rr2 += s_rw[128+f]*ev;  rr3 += s_rw[192+f]*ev;
  }
  off0[p] = o0;
  off1[p] = o1;
  r2out[0*NP + p] = 1.0f/(1.0f + __expf(-rr0));
  r2out[1*NP + p] = 1.0f/(1.0f + __expf(-rr1));
  r2out[2*NP + p] = 1.0f/(1.0f + __expf(-rr2));
  r2out[3*NP + p] = 1.0f/(1.0f + __expf(-rr3));
}

// ---------------------------------------------------------------------------
// Kernel 3: offset-shifted bilinear sampling -> fea0 (31, 256, 256)
// ---------------------------------------------------------------------------
__global__ void k_bilinear(const int* scale_p, const float* xin,
                           const float* off0, const float* off1, float* fea0) {
  int idx = blockIdx.x * blockDim.x + threadIdx.x;
  int c = idx >> 16;
  int p = idx & (NP-1);
  int py = p >> 8, px = p & 255;
  float fs = (float)(*scale_p);
  float ix = ((float)px + 0.5f)/fs - 0.5f + off0[p];
  float iy = ((float)py + 0.5f)/fs - 0.5f + off1[p];
  float x0 = floorf(ix), y0 = floorf(iy);
  float wx1 = ix - x0, wy1 = iy - y0;
  const float* img = xin + c*(HIN*WIN);
  float acc = 0.f;
  #pragma unroll
  for (int dy = 0; dy < 2; ++dy)
    #pragma unroll
    for (int dx = 0; dx < 2; ++dx) {
      float xc = x0 + (float)dx, yc = y0 + (float)dy;
      float wgt = (dx ? wx1 : 1.f - wx1) * (dy ? wy1 : 1.f - wy1);
      bool valid = (xc >= 0.f) && (xc <= (float)(WIN-1)) && (yc >= 0.f) && (yc <= (float)(HIN-1));
      int xi = (int)fminf(fmaxf(xc, 0.f), (float)(WIN-1));
      int yi = (int)fminf(fmaxf(yc, 0.f), (float)(HIN-1));
      acc += img[yi*WIN + xi] * (valid ? wgt : 0.f);
    }
  fea0[idx] = acc;
}

// ---------------------------------------------------------------------------
// Kernel 4: fused expand(32,1,5,3,3) + compress(1,32,5,3,3) 3D conv, exact.
// Output tile: 4 depth x 8 x 8. Intermediate `o` staged in LDS in 8-channel
// chunks (zero outside its true domain == conv2's zero padding). t = o2 + fea.
// ---------------------------------------------------------------------------
__global__ void __launch_bounds__(256) k_conv(const float* fea0, const float* wewt,
                                              const float* wcwt, float* tbuf) {
  __shared__ float s_fea[12*12*12];   // fea halo [d][y][x]
  __shared__ float s_o[8*8*10*10];    // o chunk [kl][d][y][x]
  __shared__ float s_we[KCH*45];
  __shared__ float s_wc[KCH*45];
  int tid = threadIdx.x;
  int bz  = blockIdx.x >> 10;
  int rem = blockIdx.x & 1023;
  int by = rem >> 5, bx = rem & 31;
  int d0 = bz*4, y0 = by*8, x0 = bx*8;
  for (int i = tid; i < KCH*45; i += 256) { s_we[i] = wewt[i]; s_wc[i] = wcwt[i]; }
  for (int i = tid; i < 1728; i += 256) {
    int ld = i / 144, r = i % 144;
    int ly = r / 12, lx = r % 12;
    int gd = d0 - 4 + ld, gy = y0 - 2 + ly, gx = x0 - 2 + lx;
    float v = 0.f;
    if (gd >= 0 && gd < CIN && gy >= 0 && gy < HO && gx >= 0 && gx < WO)
      v = fea0[gd*NP + gy*WO + gx];
    s_fea[i] = v;
  }
  int dl = tid >> 6, yl = (tid >> 3) & 7, xl = tid & 7;  // this thread's output
  float acc = 0.f;
  for (int kc = 0; kc < 4; ++kc) {
    __syncthreads();
    // expand conv for channels [kc*8, kc*8+8) into LDS
    for (int i = tid; i < 6400; i += 256) {
      int kl = i / 800, r = i % 800;
      int ld = r / 100, rr = r % 100;
      int ly = rr / 10, lx = rr % 10;
      int k  = kc*8 + kl;
      int od = d0 - 2 + ld, oy = y0 - 1 + ly, ox = x0 - 1 + lx;
      float v = 0.f;
      if (od >= 0 && od < CIN && oy >= 0 && oy < HO && ox >= 0 && ox < WO) {
        const float* wk = &s_we[k*45];
        #pragma unroll
        for (int qd = 0; qd < 5; ++qd)
          #pragma unroll
          for (int qy = 0; qy < 3; ++qy)
            #pragma unroll
            for (int qx = 0; qx < 3; ++qx)
              v += wk[qd*9 + qy*3 + qx] * s_fea[(ld+qd)*144 + (ly+qy)*12 + (lx+qx)];
      }
      s_o[i] = v;
    }
    __syncthreads();
    // compress conv: reduce this 8-channel chunk into acc
    for (int kl = 0; kl < 8; ++kl) {
      const float* wk = &s_wc[(kc*8 + kl)*45];
      const float* obase = &s_o[kl*800];
      #pragma unroll
      for (int qd = 0; qd < 5; ++qd)
        #pragma unroll
        for (int qy = 0; qy < 3; ++qy)
          #pragma unroll
          for (int qx = 0; qx < 3; ++qx)
            acc += wk[qd*9+qy*3+qx] * obase[(dl+qd)*100 + (yl+qy)*10 + (xl+qx)];
    }
  }
  int d = d0 + dl;
  if (d < CIN) {
    float f = s_fea[(dl+4)*144 + (yl+2)*12 + (xl+2)];  // fea at (d, y0+yl, x0+xl)
    tbuf[d*NP + (y0+yl)*WO + (x0+xl)] = acc + f;       // t = conv2(conv1(fea)) + fea
  }
}

// ---------------------------------------------------------------------------
// Kernel 5: mix via WMMA. u[p][k=(e,j)] = r2[e][p]*t[j][p]; out = u @ M + fea0,
// M[k=(e,j)][i] = weight_2[e][i][j]. K=124->128, N=31->32, 64 pixels/block.
// ---------------------------------------------------------------------------
#define MIXPX 64
__global__ void __launch_bounds__(128) k_mix(const float* w2in, const float* r2in,
                      const float* tbuf, const float* fea0, float* out) {
  __shared__ float s_M[128*32];       // [k][i], zero-padded
  __shared__ float s_u[MIXPX*132];    // [p][k], stride 132
  int tid = threadIdx.x;
  int p0  = blockIdx.x * MIXPX;
  for (int i = tid; i < 128*32; i += 128) {
    int k = i >> 5, n = i & 31;
    float v = 0.f;
    if (k < 124 && n < 31) {
      int e = k / 31, j = k % 31;                // k = e*31 + j
      v = w2in[e*961 + n*31 + j];
    }
    s_M[i] = v;
  }
  for (int i = tid; i < MIXPX*128; i += 128) {   // consecutive tid -> consecutive p (coalesced)
    int k = i >> 6, pl = i & 63;
    float v = 0.f;
    if (k < 124) {
      int e = k / 31, j = k % 31;
      int p = p0 + pl;
      v = r2in[e*NP + p] * tbuf[j*NP + p];
    }
    s_u[pl*132 + k] = v;
  }
  __syncthreads();
  int lane = tid & 31, wave = tid >> 5;
  int half = lane >> 4, mrow = lane & 15;
  int m0 = wave * 16;
  v8f acc0 = {}, acc1 = {};
  const float* arow = &s_u[(m0 + mrow)*132 + 2*half];
  #pragma unroll 8
  for (int k = 0; k < 128; k += 4) {
    v2f a = *(const v2f*)(arow + k);
    v2f b0, b1;
    b0.x = s_M[(k     + 2*half)*32 + mrow];
    b0.y = s_M[(k + 1 + 2*half)*32 + mrow];
    b1.x = s_M[(k     + 2*half)*32 + 16 + mrow];
    b1.y = s_M[(k + 1 + 2*half)*32 + 16 + mrow];
    acc0 = wmma_f32_16x16x4(a, b0, acc0);
    acc1 = wmma_f32_16x16x4(a, b1, acc1);
  }
  #pragma unroll
  for (int v = 0; v < 8; ++v) {
    int m = m0 + v + 8*half;
    int p = p0 + m;
    int i0 = mrow;
    int i1 = 16 + mrow;
    out[i0*NP + p] = acc0[v] + fea0[i0*NP + p];
    if (i1 < CIN)
      out[i1*NP + p] = acc1[v] + fea0[i1*NP + p];
  }
}

// ---------------------------------------------------------------------------
extern "C" void kernel_launch(void* const* d_in, const int* in_sizes, int n_in,
                              void* d_out, int out_size, void* d_ws, size_t ws_size,
                              hipStream_t stream) {
  (void)in_sizes; (void)n_in; (void)out_size; (void)ws_size;
  const float* x        = (const float*)d_in[0];
  const int*   scale_p  = (const int*)d_in[1];
  const float* we_in    = (const float*)d_in[2];
  const float* wc_in    = (const float*)d_in[3];
  const float* w2in     = (const float*)d_in[4];
  const float* body_w1  = (const float*)d_in[5];
  const float* body_b1  = (const float*)d_in[6];
  const float* body_w2  = (const float*)d_in[7];
  const float* body_b2  = (const float*)d_in[8];
  const float* r2w      = (const float*)d_in[9];
  const float* r2b      = (const float*)d_in[10];
  const float* offw     = (const float*)d_in[11];
  const float* offb     = (const float*)d_in[12];
  const float* lin1w    = (const float*)d_in[13];
  const float* lin1b    = (const float*)d_in[14];
  const float* lin2w    = (const float*)d_in[15];
  const float* lin2b    = (const float*)d_in[16];

  float* ws   = (float*)d_ws;           // ~17.8 MB used
  float* off0 = ws;                     // NP
  float* off1 = ws + NP;                // NP
  float* r2m  = ws + 2*NP;              // 4*NP
  float* fea0 = ws + 6*NP;              // 31*NP
  float* tbuf = ws + 37*NP;             // 31*NP
  float* wewt = ws + 68*NP;             // 1440
  float* wcwt = ws + 68*NP + KCH*45;    // 1440
  float* out  = (float*)d_out;

  k_prep<<<1, 256, 0, stream>>>(scale_p, we_in, wc_in, lin1w, lin1b, lin2w, lin2b, wewt, wcwt);
  k_mlp<<<NP/MLP_PX, MLP_PX, 0, stream>>>(scale_p, body_w1, body_b1, body_w2, body_b2,
                                          r2w, r2b, offw, offb, off0, off1, r2m);
  k_bilinear<<<(CIN*NP)/256, 256, 0, stream>>>(scale_p, x, off0, off1, fea0);
  k_conv<<<8*32*32, 256, 0, stream>>>(fea0, wewt, wcwt, tbuf);
  k_mix<<<NP/MIXPX, 128, 0, stream>>>(w2in, r2m, tbuf, fea0, out);
}